// LocalGrouper_76622216561216
// MI455X (gfx1250) — compile-verified
//
#include <hip/hip_runtime.h>

// Problem constants (from the reference)
#define Bn 8
#define Nn 4096
#define Dn 64
#define Sn 1024
#define Kn 32
#define R2f 0.04f          // RADIUS^2 = 0.2^2
#define OUTC 131           // D + 3 + D

typedef __attribute__((ext_vector_type(2))) float v2f;
typedef __attribute__((ext_vector_type(8))) float v8f;

// ---------------------------------------------------------------------------
// Kernel 1: farthest point sampling. One 1024-thread block per batch.
// Points stay register-resident (4 per thread). Per iteration:
// update min-distance, wave32 shuffle argmax, 32-entry LDS cross-wave argmax.
// Records fps_idx[i] = farthest BEFORE the update (matches lax.scan output),
// argmax ties break to the lowest index (matches jnp.argmax).
// ---------------------------------------------------------------------------
__global__ __launch_bounds__(1024) void fps_kernel(const float* __restrict__ xyz,
                                                   float* __restrict__ new_xyz,
                                                   int* __restrict__ fps_idx) {
  const int b    = blockIdx.x;
  const int tid  = threadIdx.x;
  const int lane = tid & 31;
  const int wid  = tid >> 5;

  __shared__ float red_v[32];
  __shared__ int   red_i[32];
  __shared__ int   s_far;

  const float* xb = xyz + (size_t)b * Nn * 3;

  float dist[4], px[4], py[4], pz[4];
#pragma unroll
  for (int i = 0; i < 4; ++i) {
    int p = tid + i * 1024;
    dist[i] = 1e10f;
    px[i] = xb[p * 3 + 0];
    py[i] = xb[p * 3 + 1];
    pz[i] = xb[p * 3 + 2];
  }

  int far = 0;
  for (int it = 0; it < Sn; ++it) {
    // record current farthest (previous iteration's argmax)
    if (tid < 3) new_xyz[((size_t)b * Sn + it) * 3 + tid] = xb[far * 3 + tid];
    if (tid == 0) fps_idx[b * Sn + it] = far;

    const float cx = xb[far * 3 + 0];
    const float cy = xb[far * 3 + 1];
    const float cz = xb[far * 3 + 2];

    float bv = -1.0f;
    int   bi = 0;
#pragma unroll
    for (int i = 0; i < 4; ++i) {
      int p = tid + i * 1024;
      float dx = px[i] - cx, dy = py[i] - cy, dz = pz[i] - cz;
      float d  = dx * dx + dy * dy + dz * dz;
      float nd = fminf(dist[i], d);
      dist[i] = nd;
      if (nd > bv) { bv = nd; bi = p; }   // strict > keeps lowest index (p ascending)
    }
    // wave32 argmax reduction
#pragma unroll
    for (int off = 16; off > 0; off >>= 1) {
      float ov = __shfl_down(bv, off, 32);
      int   oi = __shfl_down(bi, off, 32);
      if (ov > bv || (ov == bv && oi < bi)) { bv = ov; bi = oi; }
    }
    if (lane == 0) { red_v[wid] = bv; red_i[wid] = bi; }
    __syncthreads();
    if (wid == 0) {
      bv = red_v[lane];
      bi = red_i[lane];
#pragma unroll
      for (int off = 16; off > 0; off >>= 1) {
        float ov = __shfl_down(bv, off, 32);
        int   oi = __shfl_down(bi, off, 32);
        if (ov > bv || (ov == bv && oi < bi)) { bv = ov; bi = oi; }
      }
      if (lane == 0) s_far = bi;
    }
    __syncthreads();
    far = s_far;
  }
}

// ---------------------------------------------------------------------------
// Kernel 2: ball query via V_WMMA_F32_16X16X4_F32. One wave (32 threads)
// owns 16 centroids. A-fragment (16x4, K-padded with 0): lanes 0-15 hold
// (x,y) = K0,K1; lanes 16-31 hold (z,0) = K2,K3 (ISA 7.12.2). B-fragment
// mirrored for each 16-point tile. The C tile gives dot products:
// VGPR r, lanes 0-15 -> (M=r, N=lane), lanes 16-31 -> (M=r+8, N=lane-16),
// so one ballot over "dist <= r^2" yields two 16-bit half-masks.
// 256 WMMA tiles cover N=4096; masks -> LDS; first-K selection by
// popcount + wave prefix scan (equivalent to reference sort/truncate/fill).
// ---------------------------------------------------------------------------
__global__ __launch_bounds__(32) void qbp_kernel(const float* __restrict__ xyz,
                                                 const float* __restrict__ new_xyz,
                                                 int* __restrict__ neigh) {
  const int blk = blockIdx.x;          // Bn * (Sn/16) blocks
  const int b   = blk >> 6;            // Sn/16 == 64 groups per batch
  const int g   = blk & 63;
  const unsigned lane = threadIdx.x;   // single wave32
  const int m = lane & 15;
  const int s = g * 16 + m;

  __shared__ float    cn_lds[16];
  __shared__ unsigned mask_lds[16 * 128];   // 128 u32 words per centroid (4096 bits)
  __shared__ int      neigh_lds[16 * Kn];

  const float* xb   = xyz + (size_t)b * Nn * 3;
  const float* cptr = new_xyz + ((size_t)b * Sn + s) * 3;
  const float cx = cptr[0], cy = cptr[1], cz = cptr[2];

  v2f a;
  if (lane < 16) { a.x = cx; a.y = cy; }        // K0,K1
  else           { a.x = cz; a.y = 0.0f; }      // K2,K3 (pad)
  if (lane < 16) cn_lds[m] = cx * cx + cy * cy + cz * cz;
  __syncthreads();

  float cnr[8];
  const int half = (lane >> 4) ? 8 : 0;
#pragma unroll
  for (int r = 0; r < 8; ++r) cnr[r] = cn_lds[r + half];

  for (int w = 0; w < 128; ++w) {
    const int n0 = w * 32 + m;           // tile A point (both wave halves load it)
    const int n1 = n0 + 16;              // tile B point
    const float x0 = xb[n0 * 3 + 0], y0 = xb[n0 * 3 + 1], z0 = xb[n0 * 3 + 2];
    const float x1 = xb[n1 * 3 + 0], y1 = xb[n1 * 3 + 1], z1 = xb[n1 * 3 + 2];
    const float pn0 = x0 * x0 + y0 * y0 + z0 * z0;
    const float pn1 = x1 * x1 + y1 * y1 + z1 * z1;

    v2f b0, b1;
    if (lane < 16) { b0.x = x0; b0.y = y0; b1.x = x1; b1.y = y1; }
    else           { b0.x = z0; b0.y = 0.0f; b1.x = z1; b1.y = 0.0f; }

    v8f zero = {};
    v8f c0 = __builtin_amdgcn_wmma_f32_16x16x4_f32(false, a, false, b0,
                                                   (short)0, zero, false, false);
    v8f c1 = __builtin_amdgcn_wmma_f32_16x16x4_f32(false, a, false, b1,
                                                   (short)0, zero, false, false);
#pragma unroll
    for (int r = 0; r < 8; ++r) {
      const float d0 = cnr[r] + pn0 - 2.0f * c0[r];
      const float d1 = cnr[r] + pn1 - 2.0f * c1[r];
      const unsigned m0 = __builtin_amdgcn_ballot_w32(d0 <= R2f);
      const unsigned m1 = __builtin_amdgcn_ballot_w32(d1 <= R2f);
      // word for centroid r   : bits 0-15 tileA (lanes 0-15), 16-31 tileB
      const unsigned wlo = (m0 & 0xFFFFu) | (m1 << 16);
      // word for centroid r+8 : high halves of both ballots
      const unsigned whi = (m0 >> 16) | (m1 & 0xFFFF0000u);
      if (lane == 0) {                   // ballots are wave-uniform scalars
        mask_lds[r * 128 + w]       = wlo;
        mask_lds[(r + 8) * 128 + w] = whi;
      }
    }
  }
  __syncthreads();

  // ordered first-K selection per centroid
  for (int mm = 0; mm < 16; ++mm) {
    int running = 0;
#pragma unroll 1
    for (int c = 0; c < 4; ++c) {
      if (running >= Kn) break;          // running is wave-uniform
      unsigned wd = mask_lds[mm * 128 + c * 32 + lane];
      int pc = __popc(wd);
      int pre = pc;
#pragma unroll
      for (int off = 1; off < 32; off <<= 1) {
        int v = __shfl_up(pre, off, 32);
        if ((int)lane >= off) pre += v;
      }
      const int excl  = pre - pc;
      const int total = __shfl(pre, 31, 32);
      int rank = running + excl;
      unsigned t = wd;
      while (t != 0u && rank < Kn) {
        int j = __ffs(t) - 1;
        neigh_lds[mm * Kn + rank] = (c * 32 + (int)lane) * 32 + j;
        ++rank;
        t &= t - 1u;
      }
      running += total;
    }
    __syncthreads();
    const int cnt   = running < Kn ? running : Kn;  // >=1 (centroid is its own neighbor)
    const int first = neigh_lds[mm * Kn];
    if ((int)lane >= cnt) neigh_lds[mm * Kn + lane] = first;
  }
  __syncthreads();

  for (int mm = 0; mm < 16; ++mm) {
    const int ss = g * 16 + mm;
    neigh[((size_t)b * Sn + ss) * Kn + lane] = neigh_lds[mm * Kn + lane];
  }
}

// ---------------------------------------------------------------------------
// Kernel 3: gather + concat -> new_points[B,S,K,131] =
// [points[n](64) | xyz[n](3) | points[fps_idx](64)]. One 256-thread block per
// centroid; 8 waves, each wave copies whole 131-float rows (coalesced 256B
// feature rows). Pure streaming, bandwidth-bound.
// ---------------------------------------------------------------------------
__global__ __launch_bounds__(256) void gather_kernel(const float* __restrict__ xyz,
                                                     const float* __restrict__ points,
                                                     const int* __restrict__ fps_idx,
                                                     const int* __restrict__ neigh,
                                                     float* __restrict__ out_np) {
  const int blk = blockIdx.x;          // Bn * Sn
  const int b   = blk >> 10;           // Sn == 1024
  const int s   = blk & 1023;
  const int tid = threadIdx.x;

  __shared__ float anchor[Dn];
  __shared__ int   nk[Kn];

  const float* pb = points + (size_t)b * Nn * Dn;
  const float* xb = xyz + (size_t)b * Nn * 3;

  if (tid < Dn) {
    const int a = fps_idx[b * Sn + s];
    anchor[tid] = pb[(size_t)a * Dn + tid];
  }
  if (tid < Kn) nk[tid] = neigh[((size_t)b * Sn + s) * Kn + tid];
  __syncthreads();

  const int lane = tid & 31;
  const int wid  = tid >> 5;           // 8 waves
  float* rowbase = out_np + (size_t)(b * Sn + s) * Kn * OUTC;

  for (int k = wid; k < Kn; k += 8) {
    const int n = nk[k];
    const float* pr = pb + (size_t)n * Dn;
    const float* xr = xb + (size_t)n * 3;
    float* orow = rowbase + (size_t)k * OUTC;
    for (int e = lane; e < OUTC; e += 32) {
      float v;
      if (e < Dn)            v = pr[e];
      else if (e < Dn + 3)   v = xr[e - Dn];
      else                   v = anchor[e - Dn - 3];
      orow[e] = v;
    }
  }
}

// ---------------------------------------------------------------------------
extern "C" void kernel_launch(void* const* d_in, const int* in_sizes, int n_in,
                              void* d_out, int out_size, void* d_ws, size_t ws_size,
                              hipStream_t stream) {
  const float* xyz    = (const float*)d_in[0];   // [B,N,3]
  const float* points = (const float*)d_in[1];   // [B,N,D]

  float* new_xyz    = (float*)d_out;                       // [B,S,3]
  float* new_points = (float*)d_out + (size_t)Bn * Sn * 3; // [B,S,K,131]

  int* fps_idx = (int*)d_ws;                 // B*S ints
  int* neigh   = fps_idx + Bn * Sn;          // B*S*K ints

  fps_kernel<<<Bn, 1024, 0, stream>>>(xyz, new_xyz, fps_idx);
  qbp_kernel<<<Bn * (Sn / 16), 32, 0, stream>>>(xyz, new_xyz, neigh);
  gather_kernel<<<Bn * Sn, 256, 0, stream>>>(xyz, points, fps_idx, neigh, new_points);
}